// Renderer_13563506721242
// MI455X (gfx1250) — compile-verified
//
#include <hip/hip_runtime.h>

typedef float v2f __attribute__((ext_vector_type(2)));
typedef float v8f __attribute__((ext_vector_type(8)));
typedef unsigned int u32x4 __attribute__((ext_vector_type(4)));
typedef int i32x4v __attribute__((ext_vector_type(4)));
typedef int i32x8v __attribute__((ext_vector_type(8)));

#define HN 228
#define WN 304
#define SN 304
#define NV 8192
#define MF 2048
#define FARV 100.0f
#define EPSV 1e-7f
#define NEG_INF (-__builtin_inff())

// coef layout: pre-swizzled WMMA A-operand form. One 4-face block = 64 floats:
// lane l (0..15)  -> (a,b) of A-row l    at block*64 + l*2
// lane l (16..31) -> (c,0) of A-row l-16 at block*64 + l*2
#define COEF_FLOATS (MF * 16)               // 32768 floats = 128 KB
#define COEF_BYTES (COEF_FLOATS * 4)

// ---------------------------------------------------------------------------
// Kernel 1: vertex transform (unproject -> rigid transform -> project -> NDC)
// ---------------------------------------------------------------------------
__global__ void xform_verts(const float* __restrict__ verts,
                            const float* __restrict__ Kmat,
                            const float* __restrict__ Rm,
                            const float* __restrict__ tm,
                            float* __restrict__ vnd) {
  int n = blockIdx.x * blockDim.x + threadIdx.x;
  if (n >= NV) return;
  float u = verts[n * 3 + 0], v = verts[n * 3 + 1], d = verts[n * 3 + 2];
  float fx = Kmat[0], cx = Kmat[2], fy = Kmat[4], cy = Kmat[5];
  float px = u * (float)WN, py = v * (float)HN;
  float xc = (px - cx) / fx * d;
  float yc = (py - cy) / fy * d;
  float zc = d;
  float X = Rm[0] * xc + Rm[1] * yc + Rm[2] * zc + tm[0];
  float Y = Rm[3] * xc + Rm[4] * yc + Rm[5] * zc + tm[1];
  float Z = Rm[6] * xc + Rm[7] * yc + Rm[8] * zc + tm[2];
  float iz = 1.0f / (Z + EPSV);
  float uu = fminf(fmaxf((fx * X * iz + cx) / (float)WN, 0.f), 1.f);
  float vv = fminf(fmaxf((fy * Y * iz + cy) / (float)HN, 0.f), 1.f);
  vnd[n * 3 + 0] = uu * ((float)WN / (float)SN) * 2.f - 1.f;
  vnd[n * 3 + 1] = vv * ((float)HN / (float)SN) * 2.f - 1.f;
  vnd[n * 3 + 2] = Z;
}

// ---------------------------------------------------------------------------
// Kernel 2: per-face affine coefficients, written directly in WMMA A-operand
// lane order so the raster loop needs zero per-lane selects.
// ---------------------------------------------------------------------------
__global__ void face_setup(const float* __restrict__ vnd,
                           const int* __restrict__ faces,
                           float* __restrict__ coef) {
  int m = blockIdx.x * blockDim.x + threadIdx.x;
  if (m >= MF) return;
  int i0 = faces[m * 3 + 0], i1 = faces[m * 3 + 1], i2 = faces[m * 3 + 2];
  float x0 = vnd[i0 * 3 + 0], y0 = vnd[i0 * 3 + 1], z0 = vnd[i0 * 3 + 2];
  float x1 = vnd[i1 * 3 + 0], y1 = vnd[i1 * 3 + 1], z1 = vnd[i1 * 3 + 2];
  float x2 = vnd[i2 * 3 + 0], y2 = vnd[i2 * 3 + 1], z2 = vnd[i2 * 3 + 2];
  float denom = (y1 - y2) * (x0 - x2) + (x2 - x1) * (y0 - y2);
  bool ok = fabsf(denom) > 1e-10f;
  float inv = 1.0f / (ok ? denom : 1.0f);
  float a0 = (y1 - y2) * inv, b0 = (x2 - x1) * inv;
  float c0 = -(a0 * x2 + b0 * y2);
  float a1 = (y2 - y0) * inv, b1 = (x0 - x2) * inv;
  float c1 = -(a1 * x2 + b1 * y2);
  float a2 = -(a0 + a1), b2 = -(b0 + b1), c2 = 1.f - c0 - c1;
  float r0 = 1.f / z0, r1 = 1.f / z1, r2 = 1.f / z2;
  float az = a0 * r0 + a1 * r1 + a2 * r2;
  float bz = b0 * r0 + b1 * r1 + b2 * r2;
  float cz = c0 * r0 + c1 * r1 + c2 * r2;
  if (!ok) {  // degenerate: force all-outside (w == -1 everywhere)
    a0 = b0 = 0.f; c0 = -1.f;
    a1 = b1 = 0.f; c1 = -1.f;
    a2 = b2 = 0.f; c2 = -1.f;
    az = bz = 0.f; cz = 1.f;
  }
  float aq[4] = {a0, a1, a2, az};
  float bq[4] = {b0, b1, b2, bz};
  float cq[4] = {c0, c1, c2, cz};
  const int blk = m >> 2;             // 4-face block
  const int r = (m & 3) * 4;          // A-matrix row base for this face
  float* base = coef + blk * 64;
#pragma unroll
  for (int q = 0; q < 4; ++q) {
    float2* lo = (float2*)(base + (r + q) * 2);         // lanes 0..15: (a,b)
    float2* hi = (float2*)(base + (16 + r + q) * 2);    // lanes 16..31: (c,0)
    *lo = make_float2(aq[q], bq[q]);
    *hi = make_float2(cq[q], 0.f);
  }
}

// ---------------------------------------------------------------------------
// Kernel 3: WMMA rasterizer with TDM-staged coefficients.
// One wave owns a 16-pixel strip; V_WMMA_F32_16X16X4_F32 evaluates
// {w0,w1,w2,inv_z} for 4 faces x 16 pixels per issue.
//   inside = min3(w0,w1,w2) >= 0
//   ivmax  = max(inside ? inv_z : -INF)            (single accumulator)
//   depth  = ivmax > 1/FAR ? 1/ivmax : FAR         (one divide per pixel)
//   sil    = (ivmax != -INF)                       (any face was inside)
// ---------------------------------------------------------------------------
__global__ __launch_bounds__(256) void raster(const float* __restrict__ coef,
                                              float* __restrict__ out) {
  extern __shared__ float scoef[];

  // --- Stage the 128 KB coefficient table into LDS with the TDM. ---
  if (threadIdx.x < 32) {  // wave 0 only (TENSORcnt is per-wave)
    unsigned long long ga = (unsigned long long)(uintptr_t)coef;
    unsigned int lds_off = (unsigned int)(uintptr_t)scoef;
    u32x4 g0;
    g0[0] = 1u;                                   // count=1 valid descriptor
    g0[1] = lds_off;                              // lds_addr
    g0[2] = (unsigned int)(ga & 0xffffffffull);   // global_addr[31:0]
    g0[3] = (unsigned int)((ga >> 32) & 0x1ffffffull) | (2u << 30);  // addr[56:32] | type=2
    i32x8v g1;
    g1[0] = (int)(2u << 16);                      // workgroup_mask=0, data_size=4B
    g1[1] = (int)((COEF_FLOATS & 0xffff) << 16);  // tensor_dim0 low16 (bits 63:48)
    g1[2] = (int)(((COEF_FLOATS >> 16) & 0xffff) |
                  (1u << 16));                    // tensor_dim0 hi16 | tensor_dim1 low16 = 1
    g1[3] = (int)((COEF_FLOATS & 0xffff) << 16);  // tensor_dim1 hi16=0 | tile_dim0 = 32768
    g1[4] = 1;                                    // tile_dim1 = 1, tile_dim2 = 0
    g1[5] = (int)COEF_FLOATS;                     // tensor_dim0_stride low32
    g1[6] = 0;
    g1[7] = 0;
    i32x4v g2 = {0, 0, 0, 0};
    i32x4v g3 = {0, 0, 0, 0};
    i32x8v g4 = {0, 0, 0, 0, 0, 0, 0, 0};
    __builtin_amdgcn_tensor_load_to_lds(g0, g1, g2, g3, g4, 0);
    __builtin_amdgcn_s_wait_tensorcnt(0);
  }
  __syncthreads();

  const int lane = threadIdx.x & 31;
  const int wid = blockIdx.x * (blockDim.x >> 5) + (threadIdx.x >> 5);
  const int STRIPS_PER_ROW = WN / 16;        // 19
  const int NSTRIP = HN * STRIPS_PER_ROW;    // 4332 (only rows surviving crop)
  if (wid >= NSTRIP) return;                 // wave-uniform guard (after barrier)

  const int row = wid / STRIPS_PER_ROW;
  const int iy = (SN - HN) + row;            // raster rows 76..303 survive flip+crop
  const int ix = (wid % STRIPS_PER_ROW) * 16 + (lane & 15);
  const float pxc = (2.f * (float)ix + 1.f - (float)SN) / (float)SN;
  const float pyc = (2.f * (float)iy + 1.f - (float)SN) / (float)SN;

  // B-matrix (4x16 f32): lanes 0-15 carry K0,K1=(x,y); lanes 16-31 carry K2,K3=(1,0)
  v2f b;
  b.x = (lane < 16) ? pxc : 1.0f;
  b.y = (lane < 16) ? pyc : 0.0f;

  // A operand comes pre-swizzled from LDS: v2f at block*64 + lane*2.
  // 32 lanes x 8B = one conflict-free sweep of all 64 LDS banks.
  const v2f* sa = (const v2f*)scoef;
  const int idx0 = lane;                     // v2f index within a 32-v2f block
  const int NITER = MF / 4;

  float ivmax = NEG_INF;

  v2f cur = sa[idx0];                        // software-pipelined LDS read
  for (int fb = 0; fb < NITER; ++fb) {
    v2f nxt = cur;
    if (fb + 1 < NITER) nxt = sa[idx0 + (fb + 1) * 32];
    v8f c = {};
    c = __builtin_amdgcn_wmma_f32_16x16x4_f32(
        /*neg_a=*/false, cur, /*neg_b=*/false, b,
        /*c_mod=*/(short)0, c, /*reuse_a=*/false, /*reuse_b=*/false);
    // lane<16: faces {4fb,4fb+1}; lane>=16: faces {4fb+2,4fb+3}
    float m1 = fminf(fminf(c[0], c[1]), c[2]);   // v_min3
    float m2 = fminf(fminf(c[4], c[5]), c[6]);
    float iv1 = (m1 >= 0.f) ? c[3] : NEG_INF;    // v_cndmask, no clamp needed
    float iv2 = (m2 >= 0.f) ? c[7] : NEG_INF;
    ivmax = fmaxf(ivmax, fmaxf(iv1, iv2));       // v_max3
    cur = nxt;
  }

  // Merge the two half-wave face groups (same pixel in lane p and p+16).
  ivmax = fmaxf(ivmax, __shfl_xor(ivmax, 16, 32));

  if (lane < 16) {
    // min over faces of 1/inv_z == 1/max(inv_z); FAR unless ivmax > 1/FAR.
    float depth = FARV;
    if (ivmax > 1.0f / FARV) depth = 1.0f / ivmax;  // one IEEE divide per pixel
    int h = (HN - 1) - row;  // vertical flip + crop
    out[h * WN + ix] = depth;
    out[HN * WN + h * WN + ix] = (ivmax != NEG_INF) ? 1.f : 0.f;
  }
}

// ---------------------------------------------------------------------------
extern "C" void kernel_launch(void* const* d_in, const int* in_sizes, int n_in,
                              void* d_out, int out_size, void* d_ws, size_t ws_size,
                              hipStream_t stream) {
  (void)in_sizes; (void)n_in; (void)out_size; (void)ws_size;
  const float* verts = (const float*)d_in[0];
  const int* faces = (const int*)d_in[1];
  const float* Kmat = (const float*)d_in[2];
  const float* Rm = (const float*)d_in[3];
  const float* tm = (const float*)d_in[4];
  float* out = (float*)d_out;

  float* vnd = (float*)d_ws;          // 8192*3 floats
  float* coef = vnd + NV * 3;         // 2048*16 floats (128 KB), A-operand order

  // Allow 128 KB dynamic LDS (host-side attribute set; capture-safe, idempotent).
  (void)hipFuncSetAttribute((const void*)raster,
                            hipFuncAttributeMaxDynamicSharedMemorySize, COEF_BYTES);

  xform_verts<<<(NV + 255) / 256, 256, 0, stream>>>(verts, Kmat, Rm, tm, vnd);
  face_setup<<<(MF + 255) / 256, 256, 0, stream>>>(vnd, faces, coef);

  const int NSTRIP = HN * (WN / 16);            // 4332 waves
  const int WPB = 8;                            // 256 threads = 8 wave32
  const int blocks = (NSTRIP + WPB - 1) / WPB;  // 542
  raster<<<blocks, 256, COEF_BYTES, stream>>>(coef, out);
}